// AUX_OVER_IVA_69595650065034
// MI455X (gfx1250) — compile-verified
//
#include <hip/hip_runtime.h>
#include <math.h>

// Problem constants (match reference setup_inputs)
#define MM 8          // mics
#define KK 2          // targets
#define TT 2048       // frames
#define FF 513        // freq bins
#define NITER 20
#define EPSR 1e-10f
#define REGC 1e-6f
#define TOLC 1e-5f
#define FGRP 32       // f-groups for deterministic r2 reduction
#define FGCH 17       // ceil(513/32)

typedef float v2f __attribute__((ext_vector_type(2)));
typedef float v8f __attribute__((ext_vector_type(8)));

// ---------- workspace layout (float offsets) ----------
static const size_t OXS   = 0;                         // planar [f][16][t]
static const size_t SXS   = (size_t)FF * 16 * TT;      // 16,809,984
static const size_t OW    = OXS + SXS;                 // W_hat [f][8][8] cpx
static const size_t SW    = (size_t)FF * 128;
static const size_t OC    = OW + SW;                   // C [f][8][8] cpx
static const size_t SC    = (size_t)FF * 128;
static const size_t OV    = OC + SC;                   // V [k][f][8][8] cpx (reg added)
static const size_t SV    = (size_t)KK * FF * 128;
static const size_t OR2P  = OV + SV;                   // r2 partials [g][k][t]
static const size_t SR2P  = (size_t)FGRP * KK * TT;
static const size_t OWT   = OR2P + SR2P;               // weights [k][t]
static const size_t SWT   = (size_t)KK * TT;
static const size_t OSC   = OWT + SWT;                 // scale [f][k] cpx
static const size_t SSC   = (size_t)FF * 4;
static const size_t ODF   = OSC + SSC;                 // per-f |dW|^2
static const size_t OOLD  = ODF + FF;                  // per-f |W_old|^2
static const size_t ODONE = OOLD + FF;                 // int done flag
static const size_t NEEDF = ODONE + 2;

// ---------- complex helpers ----------
struct cpx { float r, i; };
__device__ __forceinline__ cpx cadd(cpx a, cpx b){ return {a.r+b.r, a.i+b.i}; }
__device__ __forceinline__ cpx csub(cpx a, cpx b){ return {a.r-b.r, a.i-b.i}; }
__device__ __forceinline__ cpx cmul(cpx a, cpx b){ return {a.r*b.r - a.i*b.i, a.r*b.i + a.i*b.r}; }
__device__ __forceinline__ cpx cconj(cpx a){ return {a.r, -a.i}; }
__device__ __forceinline__ float cabs2(cpx a){ return a.r*a.r + a.i*a.i; }
__device__ __forceinline__ cpx cdiv(cpx a, cpx b){
    float inv = 1.0f / (b.r*b.r + b.i*b.i);
    return { (a.r*b.r + a.i*b.i)*inv, (a.i*b.r - a.r*b.i)*inv };
}
__device__ __forceinline__ cpx cload(const float* p){ return {p[0], p[1]}; }
__device__ __forceinline__ void cstore(float* p, cpx a){ p[0]=a.r; p[1]=a.i; }

// ---------- init done flag ----------
__global__ void k_init_done(int* done){ if (threadIdx.x==0 && blockIdx.x==0) *done = 0; }

// ---------- init W_hat: I with lower-right (M-K) block = -I ----------
__global__ void k_init_w(float* __restrict__ W){
    int idx = blockIdx.x * blockDim.x + threadIdx.x;           // over FF*64
    if (idx >= FF*64) return;
    int e = idx & 63; int m = e >> 3, n = e & 7;
    float re = (m==n) ? (m < KK ? 1.0f : -1.0f) : 0.0f;
    W[(size_t)idx*2]   = re;
    W[(size_t)idx*2+1] = 0.0f;
}

// ---------- transpose X (M,T,F,2) -> Xs [f][row16][t]  (rows 0-7 Re, 8-15 Im) ----------
__global__ void k_transpose(const float* __restrict__ X, float* __restrict__ Xs){
    __shared__ float tre[32][33];
    __shared__ float tim[32][33];
    int m  = blockIdx.z;
    int t0 = blockIdx.x * 32;
    int f0 = blockIdx.y * 32;
    int c0 = threadIdx.x & 31;
    int c1 = threadIdx.x >> 5;     // 0..7
    #pragma unroll
    for (int i = 0; i < 4; ++i){
        int tt = c1 + i*8;
        int f  = f0 + c0;
        if (f < FF){
            const float* p = X + (((size_t)m*TT + (t0+tt))*FF + f)*2;
            tre[tt][c0] = p[0];
            tim[tt][c0] = p[1];
        }
    }
    __syncthreads();
    #pragma unroll
    for (int i = 0; i < 4; ++i){
        int ff = c1 + i*8;
        int f  = f0 + ff;
        if (f < FF){
            size_t base = ((size_t)f*16 + m)*TT + t0 + c0;
            Xs[base]            = tre[c0][ff];
            Xs[base + 8u*TT]    = tim[c0][ff];
        }
    }
}

// ---------- WMMA weighted covariance: per (f,k) block computes 8x8 complex Gram ----------
// G(16x16) = S * diag(w) * S^T  via V_WMMA_F32_16X16X4_F32 chains (8 waves x 64 steps)
__global__ void k_cov_wmma(const float* __restrict__ Xs,
                           const float* __restrict__ wt,   // nullptr => weight 1
                           float* __restrict__ out,        // [gridDim.y][FF][8][8] cpx
                           const int* __restrict__ done,
                           int addReg, int checkDone)
{
    if (checkDone){
        int dn = __builtin_amdgcn_readfirstlane(done[0]);
        if (dn) return;
    }
    const int f    = blockIdx.x;
    const int k    = blockIdx.y;
    const int tid  = threadIdx.x;
    const int lane = tid & 31;
    const int wave = tid >> 5;
    const int row  = lane & 15;
    const int kb   = (lane >> 4) * 2;    // K-slot base within chunk

    const float* sbase = Xs + ((size_t)f*16 + row) * TT;
    const float* wbase = wt ? (wt + (size_t)k * TT) : nullptr;

    v8f acc = {0.f,0.f,0.f,0.f,0.f,0.f,0.f,0.f};
    const int tstart = wave * (TT / 8);
    #pragma unroll 4
    for (int c = 0; c < (TT/8)/4; ++c){
        int t0 = tstart + c*4 + kb;
        v2f s = *(const v2f*)(sbase + t0);      // A tile: S[row][t0..t0+1]
        v2f b;
        if (wbase){
            v2f w2 = *(const v2f*)(wbase + t0); // weights for these K-slots
            b = s * w2;                          // B = diag(w) * S^T (same lane layout)
        } else {
            b = s;
        }
        acc = __builtin_amdgcn_wmma_f32_16x16x4_f32(
                  false, s, false, b, (short)0, acc, false, false);
    }

    __shared__ float gbuf[8][256];
    __shared__ float gsum[256];
    #pragma unroll
    for (int r = 0; r < 8; ++r)
        gbuf[wave][(r + 8*(lane>>4))*16 + (lane&15)] = acc[r];
    __syncthreads();
    float s = 0.f;
    #pragma unroll
    for (int w = 0; w < 8; ++w) s += gbuf[w][tid];
    gsum[tid] = s;
    __syncthreads();

    if (tid < 64){
        int m = tid >> 3, n = tid & 7;
        const float invT = 1.0f / (float)TT;
        float re = (gsum[m*16+n] + gsum[(m+8)*16 + (n+8)]) * invT;
        float im = (gsum[(m+8)*16 + n] - gsum[m*16 + (n+8)]) * invT;
        if (addReg && m == n) re += REGC;
        float* o = out + (((size_t)k*FF + f)*64 + tid)*2;
        o[0] = re; o[1] = im;
    }
}

// ---------- Y + |Y|^2 partial sums over f-groups (deterministic, no atomics) ----------
__global__ void k_yr2(const float* __restrict__ Xs, const float* __restrict__ W,
                      float* __restrict__ r2p, const int* __restrict__ done)
{
    int dn = __builtin_amdgcn_readfirstlane(done[0]);
    if (dn) return;
    int t = blockIdx.x * blockDim.x + threadIdx.x;
    int g = blockIdx.y;
    int f0 = g * FGCH, f1 = f0 + FGCH; if (f1 > FF) f1 = FF;
    float a0 = 0.f, a1 = 0.f;
    for (int f = f0; f < f1; ++f){
        const float* wr = W + (size_t)f*128;
        const float* xb = Xs + (size_t)f*16*TT + t;
        float y0r=0,y0i=0,y1r=0,y1i=0;
        #pragma unroll
        for (int m = 0; m < 8; ++m){
            float a = xb[(size_t)m*TT];
            float b = xb[(size_t)(m+8)*TT];
            float w0r = wr[m*2],      w0i = wr[m*2+1];
            float w1r = wr[16+m*2],   w1i = wr[16+m*2+1];
            y0r += w0r*a - w0i*b;  y0i += w0r*b + w0i*a;
            y1r += w1r*a - w1i*b;  y1i += w1r*b + w1i*a;
        }
        a0 += y0r*y0r + y0i*y0i;
        a1 += y1r*y1r + y1i*y1i;
    }
    r2p[((size_t)g*KK + 0)*TT + t] = a0;
    r2p[((size_t)g*KK + 1)*TT + t] = a1;
}

// ---------- contrast weights: w = 1/(2r), r = sqrt(max(sum,EPSR)) ----------
__global__ void k_weights(const float* __restrict__ r2p, float* __restrict__ wt,
                          const int* __restrict__ done)
{
    int dn = __builtin_amdgcn_readfirstlane(done[0]);
    if (dn) return;
    int i = blockIdx.x * blockDim.x + threadIdx.x;
    if (i >= KK*TT) return;
    int k = i / TT, t = i % TT;
    float s = 0.f;
    for (int g = 0; g < FGRP; ++g) s += r2p[((size_t)g*KK + k)*TT + t];
    float r = sqrtf(fmaxf(s, EPSR));
    r = fmaxf(r, EPSR);
    wt[(size_t)k*TT + t] = 0.5f / r;
}

// ---------- per-frequency W update: two 8x8 pivoted complex solves + J block ----------
__global__ void k_update(const float* __restrict__ V, const float* __restrict__ Cm,
                         float* __restrict__ W, float* __restrict__ wdiff,
                         float* __restrict__ wold, const int* __restrict__ done)
{
    int dn = __builtin_amdgcn_readfirstlane(done[0]);
    if (dn) return;
    int f = blockIdx.x * blockDim.x + threadIdx.x;
    if (f >= FF) return;

    cpx Wl[8][8], Wo[8][8];
    const float* wp = W + (size_t)f*128;
    for (int m = 0; m < 8; ++m)
        for (int n = 0; n < 8; ++n){ Wl[m][n] = cload(wp + (m*8+n)*2); Wo[m][n] = Wl[m][n]; }

    for (int k = 0; k < KK; ++k){
        cpx Vl[8][8];
        const float* vp = V + (((size_t)k*FF + f)*64)*2;
        for (int m=0;m<8;++m) for (int n=0;n<8;++n) Vl[m][n] = cload(vp + (m*8+n)*2);

        cpx A[8][9];
        for (int i=0;i<8;++i){
            for (int j=0;j<8;++j){
                cpx acc = {0,0};
                for (int m=0;m<8;++m) acc = cadd(acc, cmul(Wl[i][m], Vl[m][j]));
                A[i][j] = acc;
            }
            A[i][8].r = (i==k) ? 1.0f : 0.0f; A[i][8].i = 0.0f;
        }
        // LU with partial pivoting on [A | e_k]
        for (int col = 0; col < 8; ++col){
            int p = col; float best = cabs2(A[col][col]);
            for (int rr = col+1; rr < 8; ++rr){
                float v = cabs2(A[rr][col]);
                if (v > best){ best = v; p = rr; }
            }
            if (p != col)
                for (int j = 0; j < 9; ++j){ cpx tmp = A[col][j]; A[col][j] = A[p][j]; A[p][j] = tmp; }
            for (int rr = col+1; rr < 8; ++rr){
                cpx fac = cdiv(A[rr][col], A[col][col]);
                for (int j = col; j < 9; ++j) A[rr][j] = csub(A[rr][j], cmul(fac, A[col][j]));
            }
        }
        cpx wv[8];
        for (int i = 7; i >= 0; --i){
            cpx acc = A[i][8];
            for (int j = i+1; j < 8; ++j) acc = csub(acc, cmul(A[i][j], wv[j]));
            wv[i] = cdiv(acc, A[i][i]);
        }
        // denom = sqrt(Re(w^H V w) + EPSR)
        float s = 0.f;
        for (int i = 0; i < 8; ++i){
            cpx row = {0,0};
            for (int j = 0; j < 8; ++j) row = cadd(row, cmul(Vl[i][j], wv[j]));
            s += cmul(cconj(wv[i]), row).r;
        }
        float invd = 1.0f / sqrtf(s + EPSR);
        for (int m = 0; m < 8; ++m){ Wl[k][m].r = wv[m].r * invd; Wl[k][m].i = -wv[m].i * invd; }
    }

    // J block: tmp = Wn[:2] @ C ; solve tmp[:, :2] X = tmp[:, 2:] ; Wn[2+i][k] = conj(X[k][i])
    const float* cp = Cm + (size_t)f*128;
    cpx tm[2][8];
    for (int k2 = 0; k2 < 2; ++k2)
        for (int j = 0; j < 8; ++j){
            cpx acc = {0,0};
            for (int m = 0; m < 8; ++m) acc = cadd(acc, cmul(Wl[k2][m], cload(cp + (m*8+j)*2)));
            tm[k2][j] = acc;
        }
    cpx det = csub(cmul(tm[0][0], tm[1][1]), cmul(tm[0][1], tm[1][0]));
    for (int j = 2; j < 8; ++j){
        cpx x0 = cdiv(csub(cmul(tm[1][1], tm[0][j]), cmul(tm[0][1], tm[1][j])), det);
        cpx x1 = cdiv(csub(cmul(tm[0][0], tm[1][j]), cmul(tm[1][0], tm[0][j])), det);
        Wl[j][0] = cconj(x0);
        Wl[j][1] = cconj(x1);
    }

    float d = 0.f, o = 0.f;
    for (int m = 0; m < 8; ++m)
        for (int n = 0; n < 8; ++n){
            cpx dd = csub(Wl[m][n], Wo[m][n]);
            d += cabs2(dd); o += cabs2(Wo[m][n]);
        }
    wdiff[f] = d; wold[f] = o;

    float* wpw = W + (size_t)f*128;
    for (int m = 0; m < 8; ++m)
        for (int n = 0; n < 8; ++n) cstore(wpw + (m*8+n)*2, Wl[m][n]);
}

// ---------- convergence: rel = ||dW||/max(||W_old||, eps) ----------
__global__ void k_conv(const float* __restrict__ wdiff, const float* __restrict__ wold,
                       int* __restrict__ done)
{
    if (threadIdx.x != 0 || blockIdx.x != 0) return;
    if (done[0]) return;
    float d = 0.f, o = 0.f;
    for (int f = 0; f < FF; ++f){ d += wdiff[f]; o += wold[f]; }
    float rel = sqrtf(d) / fmaxf(sqrtf(o), 1.1920929e-07f);
    if (rel < TOLC) done[0] = 1;
}

// ---------- MDP scale: row REF_MIC-1 of pinv(W) = W^H (W W^H)^-1 ----------
__global__ void k_scale(const float* __restrict__ W, float* __restrict__ sc){
    int f = blockIdx.x * blockDim.x + threadIdx.x;
    if (f >= FF) return;
    const float* wp = W + (size_t)f*128;
    cpx w0[8], w1[8];
    for (int m = 0; m < 8; ++m){ w0[m] = cload(wp + m*2); w1[m] = cload(wp + (8+m)*2); }
    cpx g00 = {0,0}, g01 = {0,0}, g11 = {0,0};
    for (int m = 0; m < 8; ++m){
        g00 = cadd(g00, cmul(w0[m], cconj(w0[m])));
        g01 = cadd(g01, cmul(w0[m], cconj(w1[m])));
        g11 = cadd(g11, cmul(w1[m], cconj(w1[m])));
    }
    cpx g10 = cconj(g01);
    cpx det = csub(cmul(g00, g11), cmul(g01, g10));
    cpx i00 = cdiv(g11, det);
    cpx i01 = cdiv(cpx{-g01.r, -g01.i}, det);
    cpx i10 = cdiv(cpx{-g10.r, -g10.i}, det);
    cpx i11 = cdiv(g00, det);
    cpx c0 = cconj(w0[0]), c1 = cconj(w1[0]);   // row m = REF_MIC-1 = 0 of W^H
    cstore(sc + (size_t)f*4,     cadd(cmul(c0, i00), cmul(c1, i10)));
    cstore(sc + (size_t)f*4 + 2, cadd(cmul(c0, i01), cmul(c1, i11)));
}

// ---------- final output: Y_out[k][t][f] = scale[f][k] * (W[f][k] . x[f][:,t]) ----------
__global__ void k_output(const float* __restrict__ X, const float* __restrict__ W,
                         const float* __restrict__ sc, float* __restrict__ out)
{
    int f = blockIdx.x * blockDim.x + threadIdx.x;
    int t = blockIdx.y;
    if (f >= FF) return;
    const float* wp = W + (size_t)f*128;
    cpx y0 = {0,0}, y1 = {0,0};
    #pragma unroll
    for (int m = 0; m < 8; ++m){
        cpx x = cload(X + (((size_t)m*TT + t)*FF + f)*2);
        y0 = cadd(y0, cmul(cload(wp + m*2), x));
        y1 = cadd(y1, cmul(cload(wp + (8+m)*2), x));
    }
    y0 = cmul(cload(sc + (size_t)f*4), y0);
    y1 = cmul(cload(sc + (size_t)f*4 + 2), y1);
    cstore(out + (((size_t)0*TT + t)*FF + f)*2, y0);
    cstore(out + (((size_t)1*TT + t)*FF + f)*2, y1);
}

__global__ void k_zero_out(float* out, int n){
    int i = blockIdx.x * blockDim.x + threadIdx.x;
    if (i < n) out[i] = 0.f;
}

extern "C" void kernel_launch(void* const* d_in, const int* in_sizes, int n_in,
                              void* d_out, int out_size, void* d_ws, size_t ws_size,
                              hipStream_t stream)
{
    const float* X = (const float*)d_in[0];
    float* out = (float*)d_out;

    if (ws_size < NEEDF * sizeof(float)){
        // workspace too small: emit defined output and bail
        k_zero_out<<<(out_size + 255)/256, 256, 0, stream>>>(out, out_size);
        return;
    }

    float* ws    = (float*)d_ws;
    float* Xs    = ws + OXS;
    float* Wbuf  = ws + OW;
    float* Cbuf  = ws + OC;
    float* Vbuf  = ws + OV;
    float* r2p   = ws + OR2P;
    float* wtb   = ws + OWT;
    float* scb   = ws + OSC;
    float* wdiff = ws + ODF;
    float* woldb = ws + OOLD;
    int*   done  = (int*)(ws + ODONE);

    k_init_done<<<1, 1, 0, stream>>>(done);
    k_init_w<<<(FF*64 + 255)/256, 256, 0, stream>>>(Wbuf);
    k_transpose<<<dim3(TT/32, (FF+31)/32, MM), 256, 0, stream>>>(X, Xs);

    // C = X X^H / T  (unweighted, no reg, never skipped)
    k_cov_wmma<<<dim3(FF, 1), 256, 0, stream>>>(Xs, nullptr, Cbuf, done, 0, 0);

    for (int it = 0; it < NITER; ++it){
        k_yr2<<<dim3(TT/256, FGRP), 256, 0, stream>>>(Xs, Wbuf, r2p, done);
        k_weights<<<(KK*TT + 255)/256, 256, 0, stream>>>(r2p, wtb, done);
        k_cov_wmma<<<dim3(FF, KK), 256, 0, stream>>>(Xs, wtb, Vbuf, done, 1, 1);
        k_update<<<(FF + 63)/64, 64, 0, stream>>>(Vbuf, Cbuf, Wbuf, wdiff, woldb, done);
        k_conv<<<1, 1, 0, stream>>>(wdiff, woldb, done);
    }

    k_scale<<<(FF + 127)/128, 128, 0, stream>>>(Wbuf, scb);
    k_output<<<dim3((FF + 255)/256, TT), 256, 0, stream>>>(X, Wbuf, scb, out);
}